// Next_Node_Probability_Calculator_for_group_35493609734351
// MI455X (gfx1250) — compile-verified
//
#include <hip/hip_runtime.h>
#include <hip/hip_bf16.h>
#include <math.h>

// ---------------------------------------------------------------------------
// Fused pointer-attention decoder for MI455X (gfx1250, wave32).
// One wave (32 threads) processes 16 rows of G for one batch through the
// whole pipeline using V_WMMA_F32_16X16X4_F32 for every matmul stage.
// Stage-1 concat GEMM is split into three straight-line loops (one per
// concat region) so the hot 8-WMMA bursts have no per-iteration branching.
// ---------------------------------------------------------------------------

typedef float v2f __attribute__((ext_vector_type(2)));
typedef float v8f __attribute__((ext_vector_type(8)));

__device__ __forceinline__ v8f wmma4(v2f a, v2f b, v8f c) {
  // D = A(16x4) * B(4x16) + C(16x16), fp32
  return __builtin_amdgcn_wmma_f32_16x16x4_f32(
      /*neg_a=*/false, a, /*neg_b=*/false, b,
      /*c_mod=*/(short)0, c, /*reuse_a=*/false, /*reuse_b=*/false);
}

__device__ __forceinline__ float half16_max(float x) {
  x = fmaxf(x, __shfl_xor(x, 1));
  x = fmaxf(x, __shfl_xor(x, 2));
  x = fmaxf(x, __shfl_xor(x, 4));
  x = fmaxf(x, __shfl_xor(x, 8));
  return x;
}
__device__ __forceinline__ float half16_sum(float x) {
  x += __shfl_xor(x, 1);
  x += __shfl_xor(x, 2);
  x += __shfl_xor(x, 4);
  x += __shfl_xor(x, 8);
  return x;
}

__global__ __launch_bounds__(32) void fused_pointer_decoder(
    const float* __restrict__ last,   // [B,100,128]
    const float* __restrict__ first,  // [B,100,128]
    const float* __restrict__ gfeat,  // [B,128]
    const float* __restrict__ kmat,   // [B,8,100,16]
    const float* __restrict__ vmat,   // [B,8,100,16]
    const float* __restrict__ shk,    // [B,128,100]
    const float* __restrict__ mask,   // [B,100,100]
    const float* __restrict__ Wq,     // [128,384]
    const float* __restrict__ Wc,     // [128,128]
    const float* __restrict__ bc,     // [128]
    float* __restrict__ outp)         // [B,100,100]
{
  constexpr int G = 100, N = 100, E = 128, H = 8, KD = 16, CIN = 384;
  constexpr int NT = 7;     // 7 tiles of 16 cover N (padded to 112)
  constexpr int NP = NT * 16;

  __shared__ float sMask[16 * NP];  // masked padding = -1e30
  __shared__ float sQ[16 * E];      // q*0.25, later reused for mh_atten_out
  __shared__ float sP[16 * NP];     // softmax numerators (per head)
  __shared__ float sO[16 * E];      // per-head outputs concatenated

  const int b     = blockIdx.x / NT;
  const int gbase = (blockIdx.x % NT) * 16;
  const int lane  = threadIdx.x;        // 0..31
  const int lid   = lane & 15;          // position within 16-lane half
  const int hi    = lane >> 4;          // 0 or 1
  const int half8 = hi * 8;             // C-layout row offset for this half
  const int kk0   = hi * 2;             // A/B-layout K offset for this half

  const v8f vzero = {0.f, 0.f, 0.f, 0.f, 0.f, 0.f, 0.f, 0.f};

  // ---- Stage 0: mask tile -> LDS (pad n>=100 with -1e30) -------------------
  for (int i = lane; i < 16 * NP; i += 32) {
    const int r = i / NP, c = i % NP;
    int g = gbase + r; if (g > G - 1) g = G - 1;
    sMask[i] = (c < N) ? mask[((size_t)b * G + g) * N + c] : -1.0e30f;
  }
  __syncthreads();

  // ---- Stage 1: q[16,128] = cat(graph,first,last)[16,384] @ Wq.T ----------
  int grow = gbase + lid; if (grow > G - 1) grow = G - 1;
  // per-lane A-operand base pointers for the three concat regions
  const float* aptr0 = gfeat + (size_t)b * E + kk0;                         // cols 0..127
  const float* aptr1 = first + ((size_t)b * G + grow) * E + kk0;            // cols 128..255
  const float* aptr2 = last  + ((size_t)b * G + grow) * E + kk0;            // cols 256..383
  // per-lane B-operand base pointer into Wq (row = output feature)
  const float* wqp = Wq + (size_t)lid * CIN + kk0;

  v8f qa[8];
#pragma unroll
  for (int t = 0; t < 8; ++t) qa[t] = vzero;

  // region 0: graph feature (row-broadcast A)
  for (int k0 = 0; k0 < 128; k0 += 4) {
    v2f a; a.x = aptr0[k0]; a.y = aptr0[k0 + 1];
#pragma unroll
    for (int t = 0; t < 8; ++t) {
      const float* wp = wqp + (size_t)t * 16 * CIN + k0;
      v2f bb; bb.x = wp[0]; bb.y = wp[1];
      qa[t] = wmma4(a, bb, qa[t]);
    }
  }
  // region 1: encoded_first_node
  for (int k0 = 0; k0 < 128; k0 += 4) {
    v2f a; a.x = aptr1[k0]; a.y = aptr1[k0 + 1];
#pragma unroll
    for (int t = 0; t < 8; ++t) {
      const float* wp = wqp + (size_t)t * 16 * CIN + 128 + k0;
      v2f bb; bb.x = wp[0]; bb.y = wp[1];
      qa[t] = wmma4(a, bb, qa[t]);
    }
  }
  // region 2: encoded_LAST_NODE
  for (int k0 = 0; k0 < 128; k0 += 4) {
    v2f a; a.x = aptr2[k0]; a.y = aptr2[k0 + 1];
#pragma unroll
    for (int t = 0; t < 8; ++t) {
      const float* wp = wqp + (size_t)t * 16 * CIN + 256 + k0;
      v2f bb; bb.x = wp[0]; bb.y = wp[1];
      qa[t] = wmma4(a, bb, qa[t]);
    }
  }

  // dump q * (1/sqrt(Kd)) to LDS (row-major 16 x 128)
#pragma unroll
  for (int t = 0; t < 8; ++t)
#pragma unroll
    for (int r = 0; r < 8; ++r)
      sQ[(r + half8) * E + t * 16 + lid] = qa[t][r] * 0.25f;
  __syncthreads();

  // ---- Stage 2+3: masked multi-head attention -----------------------------
  const float* kb = kmat + (size_t)b * H * N * KD;
  const float* vb = vmat + (size_t)b * H * N * KD;

  for (int h = 0; h < H; ++h) {
    const float* kh = kb + (size_t)h * N * KD;
    const float* vh = vb + (size_t)h * N * KD;

    // score[16,112] = q_h[16,16] @ k_h^T
    v8f s[NT];
#pragma unroll
    for (int t = 0; t < NT; ++t) s[t] = vzero;
#pragma unroll
    for (int kd0 = 0; kd0 < KD; kd0 += 4) {
      v2f a;
      { const float* qp = &sQ[lid * E + h * 16 + kd0 + kk0];
        a.x = qp[0]; a.y = qp[1]; }
#pragma unroll
      for (int t = 0; t < NT; ++t) {
        int n = t * 16 + lid; if (n > N - 1) n = N - 1;  // clamp pad cols
        const float* kp = kh + n * KD + kd0 + kk0;
        v2f bb; bb.x = kp[0]; bb.y = kp[1];
        s[t] = wmma4(a, bb, s[t]);
      }
    }

    // + mask, row softmax (numerators kept, denominator applied after P@V)
    float rsum[8];
#pragma unroll
    for (int r = 0; r < 8; ++r) {
      float m = -3.0e38f;
#pragma unroll
      for (int t = 0; t < NT; ++t) {
        float e = s[t][r] + sMask[(r + half8) * NP + t * 16 + lid];
        s[t][r] = e;
        m = fmaxf(m, e);
      }
      m = half16_max(m);
      float su = 0.f;
#pragma unroll
      for (int t = 0; t < NT; ++t) {
        float p = __expf(s[t][r] - m);
        s[t][r] = p;
        su += p;
      }
      rsum[r] = half16_sum(su);
    }
    // stage P (numerators) to LDS for A-fragment re-layout
#pragma unroll
    for (int r = 0; r < 8; ++r)
#pragma unroll
      for (int t = 0; t < NT; ++t)
        sP[(r + half8) * NP + t * 16 + lid] = s[t][r];
    __syncthreads();

    // out_h[16,16] = P[16,112] @ v_h[112,16]  (pad rows contribute 0)
    v8f o = vzero;
    for (int n0 = 0; n0 < NP; n0 += 4) {
      v2f a;
      { const float* pp = &sP[lid * NP + n0 + kk0];
        a.x = pp[0]; a.y = pp[1]; }
      int n_0 = n0 + kk0;     if (n_0 > N - 1) n_0 = N - 1;
      int n_1 = n0 + kk0 + 1; if (n_1 > N - 1) n_1 = N - 1;
      v2f bb; bb.x = vh[n_0 * KD + lid]; bb.y = vh[n_1 * KD + lid];
      o = wmma4(a, bb, o);
    }
    // apply softmax denominator, store into head-concat buffer
#pragma unroll
    for (int r = 0; r < 8; ++r)
      sO[(r + half8) * E + h * 16 + lid] = o[r] / rsum[r];
    __syncthreads();
  }

  // ---- Stage 4: mh[16,128] = out_concat @ Wc.T + bc -----------------------
  v8f mh[8];
#pragma unroll
  for (int t = 0; t < 8; ++t) mh[t] = vzero;
  {
    const float* wcp = Wc + (size_t)lid * E + kk0;
    for (int c0 = 0; c0 < E; c0 += 4) {
      v2f a;
      { const float* op = &sO[lid * E + c0 + kk0];
        a.x = op[0]; a.y = op[1]; }
#pragma unroll
      for (int t = 0; t < 8; ++t) {
        const float* wp = wcp + (size_t)t * 16 * E + c0;
        v2f bb; bb.x = wp[0]; bb.y = wp[1];
        mh[t] = wmma4(a, bb, mh[t]);
      }
    }
  }
#pragma unroll
  for (int t = 0; t < 8; ++t) {
    const float bcv = bc[t * 16 + lid];
#pragma unroll
    for (int r = 0; r < 8; ++r)
      sQ[(r + half8) * E + t * 16 + lid] = mh[t][r] + bcv;   // reuse sQ
  }
  __syncthreads();

  // ---- Stage 5: pointer logits + tanh clip + softmax ----------------------
  const float* sk = shk + (size_t)b * E * N;
  v8f s2[NT];
#pragma unroll
  for (int t = 0; t < NT; ++t) s2[t] = vzero;
  for (int c0 = 0; c0 < E; c0 += 4) {
    v2f a;
    { const float* qp = &sQ[lid * E + c0 + kk0];
      a.x = qp[0]; a.y = qp[1]; }
#pragma unroll
    for (int t = 0; t < NT; ++t) {
      int n = t * 16 + lid; if (n > N - 1) n = N - 1;
      const float* kp = sk + (size_t)(c0 + kk0) * N + n;
      v2f bb; bb.x = kp[0]; bb.y = kp[N];
      s2[t] = wmma4(a, bb, s2[t]);
    }
  }

  const float invsq = 0.08838834764831845f;   // 1/sqrt(128)
  float* ob = outp + ((size_t)b * G + gbase) * N;
#pragma unroll
  for (int r = 0; r < 8; ++r) {
    const int g = gbase + r + half8;
    float vals[NT];
    float m = -3.0e38f;
#pragma unroll
    for (int t = 0; t < NT; ++t) {
      float x = s2[t][r] * invsq;
      x = fminf(fmaxf(x, -15.f), 15.f);
      const float e2 = __expf(2.f * x);
      const float th = (e2 - 1.f) / (e2 + 1.f);
      const float val = 10.f * th + sMask[(r + half8) * NP + t * 16 + lid];
      vals[t] = val;
      m = fmaxf(m, val);
    }
    m = half16_max(m);
    float su = 0.f;
#pragma unroll
    for (int t = 0; t < NT; ++t) {
      const float p = __expf(vals[t] - m);
      vals[t] = p;
      su += p;
    }
    su = half16_sum(su);
    const float inv = 1.f / su;
#pragma unroll
    for (int t = 0; t < NT; ++t) {
      const int n = t * 16 + lid;
      if (g < G && n < N) ob[(size_t)(r + half8) * N + n] = vals[t] * inv;
    }
  }
}

extern "C" void kernel_launch(void* const* d_in, const int* in_sizes, int n_in,
                              void* d_out, int out_size, void* d_ws, size_t ws_size,
                              hipStream_t stream) {
  const float* last  = (const float*)d_in[0];
  const float* first = (const float*)d_in[1];
  const float* gfeat = (const float*)d_in[2];
  const float* kmat  = (const float*)d_in[3];
  const float* vmat  = (const float*)d_in[4];
  const float* shk   = (const float*)d_in[5];
  const float* mask  = (const float*)d_in[6];
  const float* Wq    = (const float*)d_in[7];
  const float* Wc    = (const float*)d_in[8];
  const float* bc    = (const float*)d_in[9];
  float* outp = (float*)d_out;

  const int B = in_sizes[2] / 128;   // global_graph_feature is [B,128]
  const int gtiles = 7;              // ceil(100/16)
  fused_pointer_decoder<<<dim3(B * gtiles), dim3(32), 0, stream>>>(
      last, first, gfeat, kmat, vmat, shk, mask, Wq, Wc, bc, outp);
}